// LearnedRouter_9019431321969
// MI455X (gfx1250) — compile-verified
//
#include <hip/hip_runtime.h>
#include <hip/hip_bf16.h>
#include <stdint.h>

typedef __attribute__((ext_vector_type(2))) float v2f;
typedef __attribute__((ext_vector_type(8))) float v8f;
typedef int v4i __attribute__((vector_size(16)));
typedef __attribute__((address_space(1))) v4i* gptr_v4i;
typedef __attribute__((address_space(3))) v4i* lptr_v4i;

#define H_DIM 4096
#define E_DIM 64
#define KC 64
#define TS 68              // tile row stride: 64 + 4 dword pad (conflict-free b64 frags)
#define LS_STRIDE 65
#define BT 64              // tokens per block
#define NTHREADS 128       // 4 waves; each wave: 16 tokens x 64 experts
#define NCHUNK (H_DIM / KC)

#if defined(__has_builtin)
#if __has_builtin(__builtin_amdgcn_global_load_async_to_lds_b128)
#define HAVE_ASYNC_LDS 1
#endif
#if __has_builtin(__builtin_amdgcn_s_wait_asynccnt)
#define HAVE_WAIT_ASYNC 1
#endif
#endif

#ifdef HAVE_ASYNC_LDS
__device__ __forceinline__ void async_copy_b128(const float* g, const float* l) {
    // global -> LDS direct, tracked by ASYNCcnt (global_load_async_to_lds_b128)
    __builtin_amdgcn_global_load_async_to_lds_b128(
        (gptr_v4i)(uintptr_t)g,
        (lptr_v4i)(uint32_t)(uintptr_t)l,
        0, 0);
}
#endif

template <int N>
__device__ __forceinline__ void wait_asynccnt() {
#ifdef HAVE_WAIT_ASYNC
    __builtin_amdgcn_s_wait_asynccnt((unsigned short)N);
#else
    asm volatile("s_wait_asynccnt %0" ::"i"(N) : "memory");
#endif
}

__global__ __launch_bounds__(NTHREADS)
void router_wmma_kernel(const float* __restrict__ x,
                        const float* __restrict__ W,
                        float* __restrict__ out,
                        int T) {
    __shared__ float xs[2][BT * TS];      // 2 x 17.0 KB (ping-pong x tiles)
    __shared__ float ws[2][E_DIM * TS];   // 2 x 17.0 KB (ping-pong W tiles)
    __shared__ float ls[BT * LS_STRIDE];  // 16.25 KB logits
    __shared__ float smax[BT];
    __shared__ float sinv[BT];

    const int tid  = threadIdx.x;
    const int wave = tid >> 5;
    const int lane = tid & 31;
    const int half = lane >> 4;
    const int l16  = lane & 15;
    const int row0 = blockIdx.x * BT;

    // Stage one K-chunk (64 cols) of x[64 rows] and W[64 rows] into LDS buffer `buf`.
    // 1024 float4 slots each => 8 per thread; per wave: 16 async b128 instructions.
    auto load_tile = [&](int buf, int k0) {
#ifdef HAVE_ASYNC_LDS
#pragma unroll
        for (int j = 0; j < 8; ++j) {
            int idx = tid + j * NTHREADS;
            int r = idx >> 4, c4 = idx & 15;
            async_copy_b128(&x[(size_t)(row0 + r) * H_DIM + k0 + c4 * 4],
                            &xs[buf][r * TS + c4 * 4]);
        }
#pragma unroll
        for (int j = 0; j < 8; ++j) {
            int idx = tid + j * NTHREADS;
            int r = idx >> 4, c4 = idx & 15;
            async_copy_b128(&W[(size_t)r * H_DIM + k0 + c4 * 4],
                            &ws[buf][r * TS + c4 * 4]);
        }
#else
#pragma unroll
        for (int j = 0; j < 8; ++j) {
            int idx = tid + j * NTHREADS;
            int r = idx >> 4, c4 = idx & 15;
            const float4 v = *reinterpret_cast<const float4*>(
                &x[(size_t)(row0 + r) * H_DIM + k0 + c4 * 4]);
            *reinterpret_cast<float4*>(&xs[buf][r * TS + c4 * 4]) = v;
        }
#pragma unroll
        for (int j = 0; j < 8; ++j) {
            int idx = tid + j * NTHREADS;
            int r = idx >> 4, c4 = idx & 15;
            const float4 v = *reinterpret_cast<const float4*>(
                &W[(size_t)r * H_DIM + k0 + c4 * 4]);
            *reinterpret_cast<float4*>(&ws[buf][r * TS + c4 * 4]) = v;
        }
#endif
    };

    v8f acc[4] = {};

    const int aOff = (wave * 16 + l16) * TS + 2 * half;
    int bOff[4];
#pragma unroll
    for (int n = 0; n < 4; ++n)
        bOff[n] = (n * 16 + l16) * TS + 2 * half;

    load_tile(0, 0);   // prologue: tile 0 in flight

    for (int c = 0; c < NCHUNK; ++c) {
        const int buf = c & 1;
        if (c + 1 < NCHUNK) {
            load_tile(buf ^ 1, (c + 1) * KC);   // prefetch next tile (async)
#ifdef HAVE_ASYNC_LDS
            wait_asynccnt<16>();                // tile c complete (in-order), next still flying
#endif
        } else {
#ifdef HAVE_ASYNC_LDS
            wait_asynccnt<0>();
#endif
        }
        __syncthreads();                        // all waves' slices visible

        const float* xb = xs[buf];
        const float* wb = ws[buf];
#pragma unroll 4
        for (int kk = 0; kk < KC; kk += 4) {
            v2f a = *reinterpret_cast<const v2f*>(&xb[aOff + kk]);
#pragma unroll
            for (int n = 0; n < 4; ++n) {
                v2f b = *reinterpret_cast<const v2f*>(&wb[bOff[n] + kk]);
                acc[n] = __builtin_amdgcn_wmma_f32_16x16x4_f32(
                    false, a, false, b, (short)0, acc[n], false, false);
            }
        }
        __syncthreads();                        // done reading before overwrite
    }

    // ---- spill logits: C/D layout (vgpr i + 8*half -> row, n*16+l16 -> col) ----
#pragma unroll
    for (int n = 0; n < 4; ++n)
#pragma unroll
        for (int i = 0; i < 8; ++i)
            ls[(wave * 16 + i + 8 * half) * LS_STRIDE + n * 16 + l16] = acc[n][i];
    __syncthreads();

    // ---- per-token softmax stats + top-2 (one thread per token) ----
    if (tid < BT) {
        const float* lp = &ls[tid * LS_STRIDE];
        float l1 = lp[0]; int i1 = 0;
        float l2 = -1e30f; int i2 = 0;
        for (int j = 1; j < E_DIM; ++j) {
            float v = lp[j];
            if (v > l1)      { l2 = l1; i2 = i1; l1 = v; i1 = j; }
            else if (v > l2) { l2 = v; i2 = j; }
        }
        float sum = 0.0f;
        for (int j = 0; j < E_DIM; ++j)
            sum += __expf(lp[j] - l1);
        smax[tid] = l1;
        sinv[tid] = 1.0f / sum;

        // top-2 weights, p=1 normalized: e1/(e1+e2), e2/(e1+e2); e1 == 1
        float e2 = __expf(l2 - l1);
        float wn = 1.0f / (1.0f + e2);
        int tok = row0 + tid;
        float* wptr = out + (size_t)T * E_DIM;   // expert_weights region
        float* iptr = wptr + (size_t)T * 2;      // expert_indices region
        wptr[(size_t)tok * 2 + 0] = wn;
        wptr[(size_t)tok * 2 + 1] = e2 * wn;
        iptr[(size_t)tok * 2 + 0] = (float)i1;
        iptr[(size_t)tok * 2 + 1] = (float)i2;
    }
    __syncthreads();

    // ---- coalesced softmax score writes: 64 tokens x 64 experts ----
#pragma unroll
    for (int j = 0; j < 32; ++j) {
        int idx = tid + j * NTHREADS;            // 0..4095
        int r   = idx >> 6;
        int cc  = idx & 63;
        float s = __expf(ls[r * LS_STRIDE + cc] - smax[r]) * sinv[r];
        out[(size_t)row0 * E_DIM + idx] = s;
    }
}

extern "C" void kernel_launch(void* const* d_in, const int* in_sizes, int n_in,
                              void* d_out, int out_size, void* d_ws, size_t ws_size,
                              hipStream_t stream) {
    const float* x = (const float*)d_in[0];
    const float* W = (const float*)d_in[1];
    float* out = (float*)d_out;
    const int T = in_sizes[0] / H_DIM;     // 16384 tokens
    const int grid = T / BT;               // 256 workgroups
    router_wmma_kernel<<<grid, NTHREADS, 0, stream>>>(x, W, out, T);
}